// SAGE_sup_1168231104586
// MI455X (gfx1250) — compile-verified
//
#include <hip/hip_runtime.h>
#include <math.h>

typedef float v2f __attribute__((ext_vector_type(2)));
typedef float v8f __attribute__((ext_vector_type(8)));

// ---------------------------------------------------------------- CSR build

__global__ void zero_int_kernel(int* __restrict__ p, int n) {
  int i = blockIdx.x * blockDim.x + threadIdx.x;
  int stride = gridDim.x * blockDim.x;
  for (; i < n; i += stride) p[i] = 0;
}

__global__ void degree_kernel(const int* __restrict__ dst, int* __restrict__ cnt, int E) {
  int e = blockIdx.x * blockDim.x + threadIdx.x;
  if (e < E) atomicAdd(&cnt[dst[e]], 1);
}

// Single-workgroup exclusive scan of cnt[0..n) -> off[0..n], off[n] = total.
// Also seeds cursor[] = off[] for the binning pass. 50000 elements in 49
// chunks of 1024 (Hillis-Steele in LDS) -- microseconds, not on the
// critical path.
__global__ __launch_bounds__(1024)
void scan_kernel(const int* __restrict__ cnt, int* __restrict__ off,
                 int* __restrict__ cursor, int n) {
  __shared__ int tmp[1024];
  int carry = 0;
  for (int base = 0; base < n; base += 1024) {
    int i = base + threadIdx.x;
    int v = (i < n) ? cnt[i] : 0;
    tmp[threadIdx.x] = v;
    __syncthreads();
#pragma unroll
    for (int d = 1; d < 1024; d <<= 1) {
      int t = (threadIdx.x >= d) ? tmp[threadIdx.x - d] : 0;
      __syncthreads();
      tmp[threadIdx.x] += t;
      __syncthreads();
    }
    int excl = tmp[threadIdx.x] - v;        // exclusive within chunk
    if (i < n) {
      off[i] = carry + excl;
      cursor[i] = carry + excl;
    }
    int chunk_total = tmp[1023];
    __syncthreads();                        // protect tmp for next chunk
    carry += chunk_total;
  }
  if (threadIdx.x == 0) off[n] = carry;
}

__global__ void csr_fill_kernel(const int* __restrict__ src, const int* __restrict__ dst,
                                int* __restrict__ cursor, int* __restrict__ csr, int E) {
  int e = blockIdx.x * blockDim.x + threadIdx.x;
  if (e < E) {
    int slot = atomicAdd(&cursor[dst[e]], 1);
    csr[slot] = src[e];
  }
}

// ------------------------------------------------- fused aggregate + GEMM
// out[row,:] = act( mean_{j in N(row)} feat[j,:] @ Wl + feat[row,:] @ Wr + b )
//
// One block = 16 output rows. Phase 1: each wave pull-aggregates its rows'
// neighbor feature rows (uniform row base per wave -> coalesced 512B/1KB
// bursts; scalar CSR walk), applies 1/max(deg,1), and writes the 16xK
// A-tiles (mean-agg and self) into padded LDS. Phase 2: WMMA f32 16x16x4
// K-loops against Wl then Wr, bias preloaded in the accumulator.
// ACT: 0 = relu, 1 = sigmoid.
template<int K, int NCOL, int WAVES, int TILES, int ACT>
__global__ __launch_bounds__(WAVES * 32)
void sage_layer_kernel(const float* __restrict__ feat,
                       const int* __restrict__ off,
                       const int* __restrict__ csr,
                       const float* __restrict__ Wl,
                       const float* __restrict__ Wr,
                       const float* __restrict__ bias,
                       float* __restrict__ out, int nrows) {
  static_assert(WAVES * TILES * 16 == NCOL, "tiling must cover NCOL");
  static_assert(K % 128 == 0, "K must be a multiple of 128");
  constexpr int KP = K + 4;        // +4 float pad: fragment reads hit distinct banks
  constexpr int CH = K / 128;      // float4 chunks per lane per row
  constexpr int RPW = 16 / WAVES;  // rows aggregated per wave
  __shared__ float sA[16 * KP];    // mean-aggregated tile
  __shared__ float sX[16 * KP];    // self-feature tile

  const int rowbase = blockIdx.x * 16;
  const int lane = threadIdx.x & 31;
  const int wave = threadIdx.x >> 5;

  // ---- Phase 1: pull-mode neighbor aggregation into LDS -------------------
  for (int rr = 0; rr < RPW; ++rr) {
    const int r = wave * RPW + rr;
    const int row = rowbase + r;
    float4 acc[CH] = {};
    float4 self[CH] = {};
    float scale = 1.0f;
    if (row < nrows) {
      const int beg = off[row];
      const int end = off[row + 1];
      scale = 1.0f / fmaxf((float)(end - beg), 1.0f);
#pragma unroll
      for (int c = 0; c < CH; ++c)
        self[c] = ((const float4*)(feat + (long)row * K))[lane + 32 * c];
      for (int p = beg; p < end; ++p) {
        const int s = csr[p];                     // uniform across the wave
        if (p + 1 < end)                          // hide gather latency
          __builtin_prefetch(feat + (long)csr[p + 1] * K, 0, 0);
        const float4* rp = (const float4*)(feat + (long)s * K);
#pragma unroll
        for (int c = 0; c < CH; ++c) {
          float4 v = rp[lane + 32 * c];           // coalesced 512B burst
          acc[c].x += v.x; acc[c].y += v.y; acc[c].z += v.z; acc[c].w += v.w;
        }
      }
    }
#pragma unroll
    for (int c = 0; c < CH; ++c) {
      const int col = 4 * (lane + 32 * c);
      float4 a;
      a.x = acc[c].x * scale; a.y = acc[c].y * scale;
      a.z = acc[c].z * scale; a.w = acc[c].w * scale;
      *(float4*)&sA[r * KP + col] = a;            // KP % 4 == 0 -> aligned
      *(float4*)&sX[r * KP + col] = self[c];
    }
  }
  __syncthreads();

  // ---- Phase 2: WMMA f32 16x16x4 ------------------------------------------
  const int m = lane & 15;           // A-frag: lane encodes M
  const int khalf = (lane >> 4) * 2; // lane half selects K pair {0,1} vs {2,3}

  v8f acc[TILES];
  int ncol[TILES];
#pragma unroll
  for (int t = 0; t < TILES; ++t) {
    ncol[t] = (wave * TILES + t) * 16 + (lane & 15);
    float bv = bias[ncol[t]];        // C/D layout: all 8 VGPRs share N
#pragma unroll
    for (int j = 0; j < 8; ++j) acc[t][j] = bv;
  }

  // mean-agg @ Wl
#pragma unroll 2
  for (int k0 = 0; k0 < K; k0 += 4) {
    const int ka = k0 + khalf;
    v2f a;
    a[0] = sA[m * KP + ka];
    a[1] = sA[m * KP + ka + 1];
#pragma unroll
    for (int t = 0; t < TILES; ++t) {
      v2f b;
      b[0] = Wl[(long)ka * NCOL + ncol[t]];
      b[1] = Wl[(long)(ka + 1) * NCOL + ncol[t]];
      acc[t] = __builtin_amdgcn_wmma_f32_16x16x4_f32(
          false, a, false, b, (short)0, acc[t], false, false);
    }
  }
  // self @ Wr
#pragma unroll 2
  for (int k0 = 0; k0 < K; k0 += 4) {
    const int ka = k0 + khalf;
    v2f a;
    a[0] = sX[m * KP + ka];
    a[1] = sX[m * KP + ka + 1];
#pragma unroll
    for (int t = 0; t < TILES; ++t) {
      v2f b;
      b[0] = Wr[(long)ka * NCOL + ncol[t]];
      b[1] = Wr[(long)(ka + 1) * NCOL + ncol[t]];
      acc[t] = __builtin_amdgcn_wmma_f32_16x16x4_f32(
          false, a, false, b, (short)0, acc[t], false, false);
    }
  }

  // Store: VGPR r -> M = r + 8*(lane>>4), N = lane%16 (+ tile base).
  const int rowhalf = (lane >> 4) * 8;
#pragma unroll
  for (int t = 0; t < TILES; ++t) {
#pragma unroll
    for (int r = 0; r < 8; ++r) {
      int row = rowbase + rowhalf + r;
      if (row < nrows) {
        float v = acc[t][r];
        if (ACT == 0) {
          v = fmaxf(v, 0.0f);
        } else {
          v = 1.0f / (1.0f + __expf(-v));
        }
        out[(long)row * NCOL + ncol[t]] = v;
      }
    }
  }
}

// ---------------------------------------------------------------- launcher

extern "C" void kernel_launch(void* const* d_in, const int* in_sizes, int n_in,
                              void* d_out, int out_size, void* d_ws, size_t ws_size,
                              hipStream_t stream) {
  constexpr int FIN = 128, FHID = 256, FOUT = 64;

  const float* x   = (const float*)d_in[0];
  const int*   ei  = (const int*)d_in[1];
  const float* Wl1 = (const float*)d_in[2];
  const float* Wr1 = (const float*)d_in[3];
  const float* b1  = (const float*)d_in[4];
  const float* Wl2 = (const float*)d_in[5];
  const float* Wr2 = (const float*)d_in[6];
  const float* b2  = (const float*)d_in[7];
  float* out = (float*)d_out;

  const int N = in_sizes[0] / FIN;
  const int E = in_sizes[1] / 2;
  const int* src = ei;       // edge_index[0, :]
  const int* dst = ei + E;   // edge_index[1, :]

  // Workspace: [cnt: N][off: N+1][cursor: N][csr: E] ints, then [h: N*FHID] f32
  int* cnt    = (int*)d_ws;
  int* off    = cnt + N;
  int* cursor = off + (N + 1);
  int* csr    = cursor + N;
  float* h    = (float*)(csr + E);

  const int T = 256;
  auto blocks = [](long n, int t) { return (int)((n + t - 1) / t); };
  const int rowBlocks = (N + 15) / 16;   // 3125 for N=50000 (exact)

  // CSR build: ~1.6M int atomics total (vs 307M f32 atomics for push-scatter)
  zero_int_kernel<<<min(blocks(N, T), 4096), T, 0, stream>>>(cnt, N);
  degree_kernel<<<blocks(E, T), T, 0, stream>>>(dst, cnt, E);
  scan_kernel<<<1, 1024, 0, stream>>>(cnt, off, cursor, N);
  csr_fill_kernel<<<blocks(E, T), T, 0, stream>>>(src, dst, cursor, csr, E);

  // Layer 1: fused gather-mean + (agg@Wl1 + x@Wr1 + b1) + relu -> h
  sage_layer_kernel<FIN, FHID, 8, 2, 0>
      <<<rowBlocks, 256, 0, stream>>>(x, off, csr, Wl1, Wr1, b1, h, N);
  // Layer 2: fused gather-mean + (agg@Wl2 + h@Wr2 + b2) + sigmoid -> out
  sage_layer_kernel<FHID, FOUT, 4, 1, 1>
      <<<rowBlocks, 128, 0, stream>>>(h, off, csr, Wl2, Wr2, b2, out, N);
}